// TokenDispatcher_41893111005469
// MI455X (gfx1250) — compile-verified
//
#include <hip/hip_runtime.h>
#include <stdint.h>

// MoE token dispatch for MI455X (gfx1250): pure streaming data movement.
// Traffic: 512 MB read + 576 MB write => ~46 us floor at 23.3 TB/s HBM.
// No FLOPs, so the kernel uses the CDNA5 async global<->LDS DMA path
// (ASYNCcnt) with b128 accesses and non-temporal cache hints (zero reuse;
// the 1 GB stream would only pollute the 192 MB L2). One 8 KB output row
// per 256-thread block.

#define IN_LEN   65536
#define OUT_LEN  73728
#define NSPLITS  64
#define ALIGN_M  128
#define DDIM     2048   // floats per row (8 KB)

typedef float vfloat4 __attribute__((ext_vector_type(4)));

// ws layout (ints): [0,64)    in_off  (exclusive cumsum of splits)
//                   [64,128)  out_off (128-aligned exclusive offsets)
//                   [128,192) splits
__global__ void moe_offsets_kernel(const int* __restrict__ splits,
                                   int* __restrict__ ws) {
    if (threadIdx.x == 0 && blockIdx.x == 0) {
        int in_off = 0;
        int out_off = 0;
        for (int e = 0; e < NSPLITS; ++e) {
            int s = splits[e];
            ws[e]               = in_off;
            ws[NSPLITS + e]     = out_off;
            ws[2 * NSPLITS + e] = s;
            in_off += s;
            out_off = ((out_off + s + ALIGN_M - 1) / ALIGN_M) * ALIGN_M;
        }
    }
}

__global__ __launch_bounds__(256) void moe_dispatch_kernel(
    const float* __restrict__ inp,
    float* __restrict__ out,
    const int* __restrict__ ws) {
    __shared__ float smem[DDIM];            // 8 KB staging buffer per block

    const int row = (int)blockIdx.x;
    const int tid = (int)threadIdx.x;

    const int* in_off  = ws;
    const int* out_off = ws + NSPLITS;
    const int* spl     = ws + 2 * NSPLITS;

    // Largest e with out_off[e] <= row (out_off[0] == 0); uniform per block,
    // so this is a handful of broadcast scalar loads hitting WGP$/L2.
    int e = 0;
#pragma unroll
    for (int step = 32; step > 0; step >>= 1) {
        int cand = e + step;
        if (cand < NSPLITS && out_off[cand] <= row) e = cand;
    }
    const int local = row - out_off[e];
    float* dst = out + (size_t)row * DDIM;

    if (local < spl[e]) {
        // Valid token: copy the 8 KB row via async global->LDS->global DMA.
        const float* src = inp + (size_t)(in_off[e] + local) * DDIM;
        unsigned lds = (unsigned)(uintptr_t)(&smem[0]) + (unsigned)tid * 16u;
        uint64_t ga  = (uint64_t)(uintptr_t)src + (uint64_t)tid * 16u;
        uint64_t gd  = (uint64_t)(uintptr_t)dst + (uint64_t)tid * 16u;

        // 256 lanes x 16 B x 2 = 8 KB. INST_OFFSET applies to both the LDS
        // and global addresses, so one (lds, ga) pair covers both halves.
        // NT hint: streamed data, never re-read -> don't retain in caches.
        asm volatile("global_load_async_to_lds_b128 %0, %1, off th:TH_LOAD_NT"
                     :: "v"(lds), "v"(ga) : "memory");
        asm volatile("global_load_async_to_lds_b128 %0, %1, off offset:4096 th:TH_LOAD_NT"
                     :: "v"(lds), "v"(ga) : "memory");

        // Each lane re-reads exactly the LDS bytes it loaded -> no barrier
        // needed, just drain this wave's async loads.
        asm volatile("s_wait_asynccnt 0x0" ::: "memory");

        asm volatile("global_store_async_from_lds_b128 %0, %1, off th:TH_STORE_NT"
                     :: "v"(gd), "v"(lds) : "memory");
        asm volatile("global_store_async_from_lds_b128 %0, %1, off offset:4096 th:TH_STORE_NT"
                     :: "v"(gd), "v"(lds) : "memory");
        // S_ENDPGM performs an implicit wait-idle, draining the async stores;
        // LDS is not reused afterwards, so no explicit wait is needed.
    } else {
        // Alignment-padding slot: output must be zero (harness poisons d_out).
        // Non-temporal b128 stores (streaming, never re-read).
        vfloat4 z = (vfloat4){0.f, 0.f, 0.f, 0.f};
        vfloat4* p = (vfloat4*)dst + tid;
        __builtin_nontemporal_store(z, p);        // bytes [tid*16, tid*16+16)
        __builtin_nontemporal_store(z, p + 256);  // bytes [tid*16+4096, ...)
    }
}

extern "C" void kernel_launch(void* const* d_in, const int* in_sizes, int n_in,
                              void* d_out, int out_size, void* d_ws, size_t ws_size,
                              hipStream_t stream) {
    // setup_inputs() order: inp (f32), out (f32, unused), in_splits (i32),
    // out_splits_offsets (i32, unused).
    const float* inp       = (const float*)d_in[0];
    const int*   in_splits = (const int*)d_in[2];
    float*       out       = (float*)d_out;
    int*         ws        = (int*)d_ws;   // needs 3*64*4 = 768 bytes

    moe_offsets_kernel<<<1, 32, 0, stream>>>(in_splits, ws);
    moe_dispatch_kernel<<<OUT_LEN, 256, 0, stream>>>(inp, out, ws);
}